// MemConv_9268539425139
// MI455X (gfx1250) — compile-verified
//
#include <hip/hip_runtime.h>
#include <hip/hip_bf16.h>
#include <math.h>
#include <stdint.h>

// Problem dims (match reference)
#define Bdim 16
#define Ndim 1024
#define Fdim 256
#define Hdim 8
#define Kdim 512
#define Odim 256
#define EPSF 1e-8f

typedef __attribute__((ext_vector_type(2))) float v2f;
typedef __attribute__((ext_vector_type(8))) float v8f;

// ---------------------------------------------------------------------------
// CDNA5 async global->LDS copy (ASYNCcnt path), 16 bytes per lane.
// LDS byte address = low 32 bits of the generic pointer (aperture tag is in
// addr[63:32] per ISA 10.2); LDS address comes from the VGPR operand.
// ---------------------------------------------------------------------------
__device__ __forceinline__ void async_copy_b128(void* lds_dst, const void* gsrc) {
    uint32_t l = (uint32_t)(uintptr_t)lds_dst;
    unsigned long long g = (unsigned long long)(uintptr_t)gsrc;
    asm volatile("global_load_async_to_lds_b128 %0, %1, off"
                 :: "v"(l), "v"(g) : "memory");
}
template <int N>
__device__ __forceinline__ void wait_async() {
    asm volatile("s_wait_asynccnt %0" :: "n"(N) : "memory");
}

// ---------------------------------------------------------------------------
// Kernel 0: zero cn accumulator (B*K floats) and the kl output slot.
// ---------------------------------------------------------------------------
__global__ void zero_kernel(float* __restrict__ cn, float* __restrict__ kl_slot) {
    int idx = blockIdx.x * blockDim.x + threadIdx.x;
    if (idx < Bdim * Kdim) cn[idx] = 0.0f;
    if (idx == 0) *kl_slot = 0.0f;
}

// ---------------------------------------------------------------------------
// Kernel 0b: k2[h][k] = sum_f keys[h][k][f]^2
// ---------------------------------------------------------------------------
__global__ void k2_kernel(const float* __restrict__ keys, float* __restrict__ k2) {
    int idx = blockIdx.x * blockDim.x + threadIdx.x; // one (h,k) per thread
    if (idx >= Hdim * Kdim) return;
    const float* row = keys + (size_t)idx * Fdim;
    float s = 0.0f;
    for (int f = 0; f < Fdim; ++f) s += row[f] * row[f];
    k2[idx] = s;
}

// ---------------------------------------------------------------------------
// Kernel 1: fused scores -> C (B,N,K) + column sums cn (B,K).
// One block per (b, 16-row n-tile). 256 threads = 8 waves; each wave owns
// 4 k-tiles of 16 keys. qk via V_WMMA_F32_16X16X4_F32, K-loop over F=256.
// ---------------------------------------------------------------------------
__global__ void __launch_bounds__(256)
scores_kernel(const float* __restrict__ Q, const float* __restrict__ keys,
              const float* __restrict__ conv_w, const int* __restrict__ mask,
              const float* __restrict__ k2, float* __restrict__ C,
              float* __restrict__ cn) {
    __shared__ __align__(16) float Qs[16][Fdim + 4]; // 1040B row stride (16-aligned)
    __shared__ float acc[16][Kdim];                  // conv-weighted accumulator
    __shared__ float red[16][16];
    __shared__ float rowsumh[16];
    __shared__ float rmax[16], rsum[16];
    __shared__ float ems[16], q2s[16];

    const int tid  = threadIdx.x;
    const int w    = tid >> 5;           // wave id 0..7
    const int lane = tid & 31;
    const int m    = lane & 15;          // row (A) / col (B,D) index
    const int g    = lane >> 4;          // lane group

    const int b  = blockIdx.x >> 6;      // N/16 = 64 tiles per batch
    const int n0 = (blockIdx.x & 63) * 16;

    // Async-stage Q tile (16 x 256 = 1024 x 16B chunks, 4 per thread)
    for (int i = 0; i < 4; ++i) {
        const int idx = tid + 256 * i, rr = idx >> 6, cc = (idx & 63) * 4;
        async_copy_b128(&Qs[rr][cc],
                        &Q[((size_t)(b * Ndim + n0 + rr)) * Fdim + cc]);
    }
    if (tid < 16) ems[tid] = (float)mask[b * Ndim + n0 + tid];
    for (int i = 0; i < 32; ++i) ((float*)acc)[tid + 256 * i] = 0.0f;
    wait_async<0>();
    __syncthreads();                     // Qs visible to all waves
    if (tid < 16) {
        float s = 0.0f;
        for (int f = 0; f < Fdim; ++f) s += Qs[tid][f] * Qs[tid][f];
        q2s[tid] = s;
    }
    __syncthreads();                     // q2s visible

    for (int h = 0; h < Hdim; ++h) {
        float S[4][8];
        const float wh = conv_w[h];
        if (h + 1 < Hdim) {              // pull next head's key rows toward WGP
            const float* nk =
                keys + ((size_t)((h + 1) * Kdim + (w * 4) * 16 + m)) * Fdim;
            __builtin_prefetch(nk, 0, 1);
        }
        // ---- qk GEMM: 4 k-tiles of 16 columns per wave ----
        for (int t = 0; t < 4; ++t) {
            const int kcol = (w * 4 + t) * 16 + m; // key index for B frag / D col
            const float* kp = keys + ((size_t)(h * Kdim + kcol)) * Fdim;
            v8f c = {0.f, 0.f, 0.f, 0.f, 0.f, 0.f, 0.f, 0.f};
            for (int kk = 0; kk < Fdim; kk += 4) {
                v2f a, bb;
                a[0]  = Qs[m][kk + 2 * g];
                a[1]  = Qs[m][kk + 2 * g + 1];
                bb[0] = kp[kk + 2 * g];
                bb[1] = kp[kk + 2 * g + 1];
                c = __builtin_amdgcn_wmma_f32_16x16x4_f32(
                        false, a, false, bb, (short)0, c, false, false);
            }
            // d2 -> student-t with TAU=1 => 1/(1+max(d2,0)); em==0 => 1
            const float k2v = k2[h * Kdim + kcol];
            for (int v = 0; v < 8; ++v) {
                const int r = v + 8 * g;
                float d2 = q2s[r] + k2v - 2.0f * c[v];
                d2 = fmaxf(d2, 0.0f);
                S[t][v] = (ems[r] != 0.0f) ? (1.0f / (1.0f + d2)) : 1.0f;
            }
        }
        // ---- per-row sum over this wave's 64 keys ----
        float prow[8];
        for (int v = 0; v < 8; ++v) {
            prow[v] = S[0][v] + S[1][v] + S[2][v] + S[3][v];
            for (int off = 8; off >= 1; off >>= 1)
                prow[v] += __shfl_xor(prow[v], off, 32);
        }
        if (m == 0)
            for (int v = 0; v < 8; ++v) red[g * 8 + v][w] = prow[v];
        __syncthreads();
        if (tid < 16) {
            float s = 0.0f;
            for (int ww = 0; ww < 8; ++ww) s += red[tid][ww];
            rowsumh[tid] = s;
        }
        __syncthreads();
        // ---- accumulate conv_w[h] * em * S / rowsum into acc ----
        for (int t = 0; t < 4; ++t) {
            const int kc = (w * 4 + t) * 16 + m;
            for (int v = 0; v < 8; ++v) {
                const int r = v + 8 * g;
                acc[r][kc] += wh * ems[r] * S[t][v] / rowsumh[r];
            }
        }
        __syncthreads();
    }

    // ---- softmax over k per row, * em, write C, accumulate cn ----
    const int r = tid >> 4, c16 = tid & 15;
    float mx = -1e30f;
    for (int i = 0; i < 32; ++i) mx = fmaxf(mx, acc[r][c16 + 16 * i]);
    red[r][c16] = mx;
    __syncthreads();
    if (c16 == 0) {
        float t0 = red[r][0];
        for (int i = 1; i < 16; ++i) t0 = fmaxf(t0, red[r][i]);
        rmax[r] = t0;
    }
    __syncthreads();
    float ls = 0.0f;
    for (int i = 0; i < 32; ++i) ls += __expf(acc[r][c16 + 16 * i] - rmax[r]);
    red[r][c16] = ls;
    __syncthreads();
    if (c16 == 0) {
        float t0 = 0.0f;
        for (int i = 0; i < 16; ++i) t0 += red[r][i];
        rsum[r] = t0;
    }
    __syncthreads();
    const float emr = ems[r], inv = 1.0f / rsum[r], mxr = rmax[r];
    for (int i = 0; i < 32; ++i) {
        const int k = c16 + 16 * i;
        const float val = __expf(acc[r][k] - mxr) * inv * emr;
        acc[r][k] = val; // reuse acc for cn reduction
        C[((size_t)(b * Ndim + n0 + r)) * Kdim + k] = val;
    }
    __syncthreads();
    // one atomic per k column per block (sum of 16 rows)
    for (int kk = tid; kk < Kdim; kk += 256) {
        float s = 0.0f;
        for (int rr = 0; rr < 16; ++rr) s += acc[rr][kk];
        atomicAdd(&cn[b * Kdim + kk], s);
    }
}

// ---------------------------------------------------------------------------
// Kernel 2: KL regularizer. One block per (b,n) row; atomicAdd scalar.
// ---------------------------------------------------------------------------
__global__ void __launch_bounds__(128)
kl_kernel(const float* __restrict__ C, const float* __restrict__ cn,
          float* __restrict__ kl_out) {
    __shared__ float sred[128];
    const int b = blockIdx.x >> 10, n = blockIdx.x & 1023;
    const float* row = C + ((size_t)(b * Ndim + n)) * Kdim;
    const int tid = threadIdx.x;

    float lp = 0.0f;
    for (int i = 0; i < 4; ++i) {
        const int k = tid + 128 * i;
        const float cv = row[k];
        lp += cv * cv / (cn[b * Kdim + k] + EPSF);
    }
    sred[tid] = lp;
    __syncthreads();
    for (int s = 64; s > 0; s >>= 1) {
        if (tid < s) sred[tid] += sred[tid + s];
        __syncthreads();
    }
    const float pn = sred[0] + EPSF;
    __syncthreads();

    float lkl = 0.0f;
    for (int i = 0; i < 4; ++i) {
        const int k = tid + 128 * i;
        const float cv = row[k];
        const float P = (cv * cv / (cn[b * Kdim + k] + EPSF)) / pn;
        lkl += P * (__logf(P + EPSF) - __logf(cv + EPSF));
    }
    sred[tid] = lkl;
    __syncthreads();
    for (int s = 64; s > 0; s >>= 1) {
        if (tid < s) sred[tid] += sred[tid + s];
        __syncthreads();
    }
    if (tid == 0) atomicAdd(kl_out, 100.0f * sred[0]);
}

// ---------------------------------------------------------------------------
// Kernel 3: V[b] = C[b]^T @ Q[b]  (K x N)*(N x F).
// Double-buffered async-LDS staging overlapped with WMMA.
// grid = (B, K/16, F/64), block = 128 (4 waves, each one 16x16 f-subtile).
// Each thread issues 5 async B128 copies per 32-row chunk; completion is
// in-order, so s_wait_asynccnt 5 == "previous chunk landed".
// ---------------------------------------------------------------------------
__global__ void __launch_bounds__(128)
v_gemm_kernel(const float* __restrict__ C, const float* __restrict__ Q,
              float* __restrict__ V) {
    __shared__ __align__(16) float Cs[2][32][20]; // 80B row stride (16-aligned)
    __shared__ __align__(16) float Qs[2][32][68]; // 272B row stride (16-aligned)

    const int b = blockIdx.x, kt0 = blockIdx.y * 16, f0 = blockIdx.z * 64;
    const int tid = threadIdx.x, wv = tid >> 5, lane = tid & 31;
    const int m = lane & 15, g = lane >> 4;

    auto load_tiles = [&](int nn, int buf) {
        {   // 32x16 C tile: 128 x 16B chunks, 1 per thread
            const int rr = tid >> 2, cc = (tid & 3) * 4;
            async_copy_b128(&Cs[buf][rr][cc],
                            &C[((size_t)(b * Ndim + nn + rr)) * Kdim + kt0 + cc]);
        }
        for (int i = 0; i < 4; ++i) { // 32x64 Q tile: 512 chunks, 4 per thread
            const int idx = tid + 128 * i, rr = idx >> 4, cc = (idx & 15) * 4;
            async_copy_b128(&Qs[buf][rr][cc],
                            &Q[((size_t)(b * Ndim + nn + rr)) * Fdim + f0 + cc]);
        }
    };

    load_tiles(0, 0);
    v8f c = {0.f, 0.f, 0.f, 0.f, 0.f, 0.f, 0.f, 0.f};
    for (int nn = 0; nn < Ndim; nn += 32) {
        const int buf = (nn >> 5) & 1;
        if (nn + 32 < Ndim) {
            load_tiles(nn + 32, buf ^ 1);
            wait_async<5>();             // current chunk's 5 copies done
        } else {
            wait_async<0>();
        }
        __syncthreads();
        for (int j = 0; j < 8; ++j) {
            const int nl = j * 4;
            v2f a, bb;
            a[0]  = Cs[buf][nl + 2 * g][m];      // A = C^T : A[k][n] = C[n][k]
            a[1]  = Cs[buf][nl + 2 * g + 1][m];
            bb[0] = Qs[buf][nl + 2 * g][wv * 16 + m];
            bb[1] = Qs[buf][nl + 2 * g + 1][wv * 16 + m];
            c = __builtin_amdgcn_wmma_f32_16x16x4_f32(
                    false, a, false, bb, (short)0, c, false, false);
        }
        __syncthreads();                 // done reading buf before it's rewritten
    }
    for (int v = 0; v < 8; ++v)
        V[((size_t)(b * Kdim + kt0 + v + 8 * g)) * Fdim + f0 + wv * 16 + m] = c[v];
}

// ---------------------------------------------------------------------------
// Kernel 4: out = leaky_relu(V @ lin_w^T + lin_b).
// grid = B*(K/16), block = 256 (8 waves, 2 o-tiles each).
// ---------------------------------------------------------------------------
__global__ void __launch_bounds__(256)
out_gemm_kernel(const float* __restrict__ V, const float* __restrict__ lin_w,
                const float* __restrict__ lin_b, float* __restrict__ out) {
    __shared__ __align__(16) float Vs[16][Fdim + 4];
    const int b = blockIdx.x >> 5, kt0 = (blockIdx.x & 31) * 16;
    const int tid = threadIdx.x, w = tid >> 5, lane = tid & 31;
    const int m = lane & 15, g = lane >> 4;

    for (int i = 0; i < 4; ++i) { // 16x256 V tile: 1024 x 16B chunks
        const int idx = tid + 256 * i, rr = idx >> 6, cc = (idx & 63) * 4;
        async_copy_b128(&Vs[rr][cc],
                        &V[((size_t)(b * Kdim + kt0 + rr)) * Fdim + cc]);
    }
    wait_async<0>();
    __syncthreads();

    for (int sub = 0; sub < 2; ++sub) {
        const int ot = w * 2 + sub;            // o-tile 0..15
        const int o  = ot * 16 + m;
        const float* lw = lin_w + (size_t)o * Fdim;
        v8f c = {0.f, 0.f, 0.f, 0.f, 0.f, 0.f, 0.f, 0.f};
        for (int ff = 0; ff < Fdim; ff += 4) {
            v2f a, bb;
            a[0]  = Vs[m][ff + 2 * g];
            a[1]  = Vs[m][ff + 2 * g + 1];
            bb[0] = lw[ff + 2 * g];            // B[f][o] = lin_w[o][f]
            bb[1] = lw[ff + 2 * g + 1];
            c = __builtin_amdgcn_wmma_f32_16x16x4_f32(
                    false, a, false, bb, (short)0, c, false, false);
        }
        const float bias = lin_b[o];
        for (int v = 0; v < 8; ++v) {
            float val = c[v] + bias;
            val = (val > 0.0f) ? val : 0.01f * val;
            out[((size_t)(b * Kdim + kt0 + v + 8 * g)) * Odim + o] = val;
        }
    }
}

// ---------------------------------------------------------------------------
extern "C" void kernel_launch(void* const* d_in, const int* in_sizes, int n_in,
                              void* d_out, int out_size, void* d_ws, size_t ws_size,
                              hipStream_t stream) {
    const float* Q      = (const float*)d_in[0];
    const float* keys   = (const float*)d_in[1];
    const float* conv_w = (const float*)d_in[2];
    const float* lin_w  = (const float*)d_in[3];
    const float* lin_b  = (const float*)d_in[4];
    const int*   mask   = (const int*)d_in[5];
    float* out = (float*)d_out;

    float* ws   = (float*)d_ws;
    float* Cbuf = ws;                                       // B*N*K = 8388608
    float* Vbuf = Cbuf + (size_t)Bdim * Ndim * Kdim;        // B*K*F = 2097152
    float* cn   = Vbuf + (size_t)Bdim * Kdim * Fdim;        // B*K   = 8192
    float* k2   = cn + Bdim * Kdim;                         // H*K   = 4096
    float* kl_slot = out + (size_t)Bdim * Kdim * Odim;      // last output elem

    zero_kernel<<<(Bdim * Kdim + 255) / 256, 256, 0, stream>>>(cn, kl_slot);
    k2_kernel<<<(Hdim * Kdim + 255) / 256, 256, 0, stream>>>(keys, k2);
    scores_kernel<<<Bdim * (Ndim / 16), 256, 0, stream>>>(Q, keys, conv_w, mask,
                                                          k2, Cbuf, cn);
    kl_kernel<<<Bdim * Ndim, 128, 0, stream>>>(Cbuf, cn, kl_slot);
    v_gemm_kernel<<<dim3(Bdim, Kdim / 16, Fdim / 64), 128, 0, stream>>>(Cbuf, Q, Vbuf);
    out_gemm_kernel<<<Bdim * (Kdim / 16), 256, 0, stream>>>(Vbuf, lin_w, lin_b, out);
}